// SharedSparseMapping_31233002177254
// MI455X (gfx1250) — compile-verified
//
#include <hip/hip_runtime.h>
#include <cmath>

typedef __attribute__((ext_vector_type(2))) float v2f;
typedef __attribute__((ext_vector_type(8))) float v8f;

#define D 256
#define LN_EPS 1e-5f

// ---------------- zero workspace ----------------
__global__ void zero_f4_kernel(float4* __restrict__ p, long long n4) {
  long long i = (long long)blockIdx.x * blockDim.x + threadIdx.x;
  long long stride = (long long)gridDim.x * blockDim.x;
  float4 z; z.x = 0.f; z.y = 0.f; z.z = 0.f; z.w = 0.f;
  for (; i < n4; i += stride) p[i] = z;
}

// ---------------- SpMM scatter: one wave32 per edge ----------------
__global__ __launch_bounds__(256) void spmm_kernel(
    const float* __restrict__ x, const int* __restrict__ rows,
    const int* __restrict__ cols, const float* __restrict__ vals,
    float* __restrict__ mapped, int nnz) {
  int wave = (int)((blockIdx.x * blockDim.x + threadIdx.x) >> 5);
  int lane = threadIdx.x & 31;
  if (wave >= nnz) return;
  int r = rows[wave];
  int c = cols[wave];
  float v = vals[wave];
  const float4* xs = (const float4*)(x + (long long)c * D);
  float* m = mapped + (long long)r * D;
#pragma unroll
  for (int j = 0; j < 2; ++j) {
    float4 d = xs[lane * 2 + j];   // lane covers floats [lane*8, lane*8+8)
    int base = lane * 8 + j * 4;
    atomicAdd(m + base + 0, v * d.x);
    atomicAdd(m + base + 1, v * d.y);
    atomicAdd(m + base + 2, v * d.z);
    atomicAdd(m + base + 3, v * d.w);
  }
}

// ---------------- fused GEMM (f32 WMMA) + bias + LayerNorm + GELU ----------------
// Block: 256 threads (8 waves). Block computes 16 rows x 256 cols.
// Wave w owns N-tiles at n0 = 32*w and 32*w+16. K loop: 64 steps of 16x16x4 f32 WMMA.
__global__ __launch_bounds__(256) void gemm_ln_gelu_kernel(
    const float* __restrict__ mapped, const float* __restrict__ W,
    const float* __restrict__ bias, const float* __restrict__ gamma,
    const float* __restrict__ beta, float* __restrict__ out, int n_tgt) {
  __shared__ float hs[16][D + 4];

  const int tid  = threadIdx.x;
  const int wave = tid >> 5;
  const int lane = tid & 31;
  const int half = lane >> 4;   // 0: lanes 0-15, 1: lanes 16-31
  const int l16  = lane & 15;

  const long long row_base = (long long)blockIdx.x * 16;
  // Clamp A-row so EXEC stays all-ones through the WMMA region.
  long long arow = row_base + l16;
  if (arow > (long long)n_tgt - 1) arow = (long long)n_tgt - 1;
  const float* __restrict__ Arow = mapped + arow * D;

  const int n0 = wave * 32;

  v8f acc0 = {0.f, 0.f, 0.f, 0.f, 0.f, 0.f, 0.f, 0.f};
  v8f acc1 = {0.f, 0.f, 0.f, 0.f, 0.f, 0.f, 0.f, 0.f};

  for (int k = 0; k < D; k += 4) {
    const int ka = k + half * 2;   // lanes 0-15: K=k,k+1 ; lanes 16-31: K=k+2,k+3
    v2f a;
    a.x = Arow[ka];
    a.y = Arow[ka + 1];
    v2f b0, b1;
    b0.x = W[(long long)ka * D + n0 + l16];
    b0.y = W[(long long)(ka + 1) * D + n0 + l16];
    b1.x = W[(long long)ka * D + n0 + 16 + l16];
    b1.y = W[(long long)(ka + 1) * D + n0 + 16 + l16];
    acc0 = __builtin_amdgcn_wmma_f32_16x16x4_f32(false, a, false, b0,
                                                 (short)0, acc0, false, false);
    acc1 = __builtin_amdgcn_wmma_f32_16x16x4_f32(false, a, false, b1,
                                                 (short)0, acc1, false, false);
  }

  // C/D layout: VGPR j, lane L -> (M = j + 8*(L>=16), N = L%16) within the tile.
#pragma unroll
  for (int j = 0; j < 8; ++j) {
    const int m = j + half * 8;
    hs[m][n0 + l16]      = acc0[j] + bias[n0 + l16];
    hs[m][n0 + 16 + l16] = acc1[j] + bias[n0 + 16 + l16];
  }
  __syncthreads();

  // LayerNorm + GELU: each 16-lane half-wave owns one of the 16 rows.
  const int r = 2 * wave + half;
  float vbuf[16];
  float sum = 0.f, sq = 0.f;
#pragma unroll
  for (int kk = 0; kk < 16; ++kk) {
    const float h = hs[r][l16 + kk * 16];
    vbuf[kk] = h;
    sum += h;
    sq  += h * h;
  }
#pragma unroll
  for (int ofs = 1; ofs < 16; ofs <<= 1) {   // xor 1,2,4,8: stays inside half-wave
    sum += __shfl_xor(sum, ofs, 32);
    sq  += __shfl_xor(sq,  ofs, 32);
  }
  const float mu   = sum * (1.0f / D);
  const float var  = sq * (1.0f / D) - mu * mu;
  const float rstd = rsqrtf(var + LN_EPS);

  const long long orow = row_base + r;
  if (orow < (long long)n_tgt) {
    float* __restrict__ o = out + orow * D;
#pragma unroll
    for (int kk = 0; kk < 16; ++kk) {
      const int c = l16 + kk * 16;
      const float g = (vbuf[kk] - mu) * rstd * gamma[c] + beta[c];
      o[c] = 0.5f * g * (1.0f + erff(g * 0.70710678118654752f));
    }
  }
}

extern "C" void kernel_launch(void* const* d_in, const int* in_sizes, int n_in,
                              void* d_out, int out_size, void* d_ws, size_t ws_size,
                              hipStream_t stream) {
  const float* x     = (const float*)d_in[0];
  const int*   rows  = (const int*)d_in[1];
  const int*   cols  = (const int*)d_in[2];
  const float* vals  = (const float*)d_in[3];
  const float* W     = (const float*)d_in[4];
  const float* bias  = (const float*)d_in[5];
  const float* gamma = (const float*)d_in[6];
  const float* beta  = (const float*)d_in[7];
  float* out = (float*)d_out;

  const int nnz   = in_sizes[1];
  const int n_tgt = out_size / D;

  float* mapped = (float*)d_ws;   // n_tgt * D floats (~41 MB)

  // 1) zero the segment-sum accumulator
  const long long n4 = ((long long)n_tgt * D) / 4;
  zero_f4_kernel<<<2048, 256, 0, stream>>>((float4*)mapped, n4);

  // 2) SpMM scatter (wave-per-edge, fp32 global atomics)
  const int waves_per_block = 256 / 32;
  const int spmm_blocks = (nnz + waves_per_block - 1) / waves_per_block;
  spmm_kernel<<<spmm_blocks, 256, 0, stream>>>(x, rows, cols, vals, mapped, nnz);

  // 3) fused GEMM (f32 WMMA) + bias + LayerNorm + GELU
  const int gemm_blocks = (n_tgt + 15) / 16;
  gemm_ln_gelu_kernel<<<gemm_blocks, 256, 0, stream>>>(mapped, W, bias, gamma,
                                                       beta, out, n_tgt);
}